// GPPT_326417514916
// MI455X (gfx1250) — compile-verified
//
#include <hip/hip_runtime.h>

#define N_NODES   20000
#define N_EDGES   320000
#define DIM       512
#define CENTERS   8
#define N_CLASSES 128

typedef __attribute__((ext_vector_type(2))) float v2f;
typedef __attribute__((ext_vector_type(8))) float v8f;

// ---------------------------------------------------------------------------
// Kernel 1: sum = h (self loop), cnt = 1.  Re-initializes scratch every call.
// ---------------------------------------------------------------------------
__global__ void gppt_init(const float4* __restrict__ h4, float4* __restrict__ sum4,
                          float* __restrict__ cnt) {
  int i = blockIdx.x * blockDim.x + threadIdx.x;   // over N*D/4 float4s
  sum4[i] = h4[i];
  if (i < N_NODES) cnt[i] = 1.0f;
}

// ---------------------------------------------------------------------------
// Kernel 2: one wave32 per edge; scatter-add h[src] row into sum[dst].
// ---------------------------------------------------------------------------
__global__ void gppt_scatter(const float* __restrict__ h, const int* __restrict__ esrc,
                             const int* __restrict__ edst, float* __restrict__ sum,
                             float* __restrict__ cnt) {
  int gid  = blockIdx.x * blockDim.x + threadIdx.x;
  int e    = gid >> 5;
  int lane = gid & 31;
  if (e >= N_EDGES) return;
  int s = esrc[e];
  int t = edst[e];
  const float4* hrow = (const float4*)(h + (size_t)s * DIM);
  float* srow = sum + (size_t)t * DIM;
#pragma unroll
  for (int j = 0; j < 4; ++j) {
    int c4 = lane + j * 32;            // 128 float4s per row, coalesced per j
    float4 x = hrow[c4];
    int col = c4 * 4;
    atomicAdd(srow + col + 0, x.x);
    atomicAdd(srow + col + 1, x.y);
    atomicAdd(srow + col + 2, x.z);
    atomicAdd(srow + col + 3, x.w);
  }
  if (lane == 0) atomicAdd(cnt + t, 1.0f);
}

// ---------------------------------------------------------------------------
// Kernel 3: hm = sum/cnt (in place); route = argmax_c hm . Ws[c]
// 256 threads = 8 waves, one wave per node; Ws staged in LDS.
// ---------------------------------------------------------------------------
__global__ void gppt_route(float* __restrict__ hm, const float* __restrict__ cnt,
                           const float* __restrict__ Ws, int* __restrict__ route) {
  __shared__ float sWs[CENTERS * DIM];           // 16 KB
  for (int i = threadIdx.x; i < CENTERS * DIM; i += 256) sWs[i] = Ws[i];
  __syncthreads();

  int lane = threadIdx.x & 31;
  int wv   = threadIdx.x >> 5;
  int node = blockIdx.x * 8 + wv;                // grid = N/8 exactly
  float inv = 1.0f / cnt[node];
  float* row = hm + (size_t)node * DIM;

  float dots[CENTERS];
#pragma unroll
  for (int c = 0; c < CENTERS; ++c) dots[c] = 0.0f;

#pragma unroll
  for (int j = 0; j < 4; ++j) {
    int c4 = lane + j * 32;
    float4 x = ((float4*)row)[c4];
    x.x *= inv; x.y *= inv; x.z *= inv; x.w *= inv;
    ((float4*)row)[c4] = x;                      // hm written back in place
    int col = c4 * 4;
#pragma unroll
    for (int c = 0; c < CENTERS; ++c) {
      const float* w = sWs + c * DIM + col;
      dots[c] += x.x * w[0] + x.y * w[1] + x.z * w[2] + x.w * w[3];
    }
  }
#pragma unroll
  for (int c = 0; c < CENTERS; ++c) {
#pragma unroll
    for (int off = 16; off > 0; off >>= 1) dots[c] += __shfl_xor(dots[c], off, 32);
  }
  if (lane == 0) {
    int best = 0; float bv = dots[0];
#pragma unroll
    for (int c = 1; c < CENTERS; ++c) { if (dots[c] > bv) { bv = dots[c]; best = c; } }
    route[node] = best;
  }
}

// ---------------------------------------------------------------------------
// Kernel 4: out[n] = Wt[route[n]] @ hm[n], via f32 WMMA 16x16x4.
// One wave per (16-node tile, 16-class block); 8 accumulators (one/expert),
// per-row select with route[] after the K loop.
// A layout (16x4 f32): lane l holds M=l&15; VGPR v -> K = k0 + v + 2*(l>=16).
// B layout (4x16 f32): lane l holds N=l&15; VGPR v -> K = k0 + v + 2*(l>=16).
// D layout: VGPR v, lane l -> row M = v + 8*(l>=16), col N = l&15.
// ---------------------------------------------------------------------------
__global__ void __launch_bounds__(128) gppt_expert(
    const float* __restrict__ hm, const float* __restrict__ Wt,
    const int* __restrict__ route, float* __restrict__ out) {
  const int lane   = threadIdx.x;                 // 0..31
  const int tile   = blockIdx.x;                  // node tile (16 nodes)
  const int cblock = blockIdx.y * 4 + threadIdx.y;// class block (16 classes)
  const int hi = lane >> 4;
  const int lo = lane & 15;
  const int node = tile * 16 + lo;                // A row
  const int cls  = cblock * 16 + lo;              // B col

  const float* aptr  = hm + (size_t)node * DIM + 2 * hi;
  const float* bbase = Wt + (size_t)cls * DIM + 2 * hi;

  v8f zero = {};
  v8f acc[CENTERS];
#pragma unroll
  for (int c = 0; c < CENTERS; ++c) acc[c] = zero;

  for (int k0 = 0; k0 < DIM; k0 += 4) {
    v2f a = *(const v2f*)(aptr + k0);
#pragma unroll
    for (int c = 0; c < CENTERS; ++c) {
      v2f b = *(const v2f*)(bbase + (size_t)c * (N_CLASSES * DIM) + k0);
      acc[c] = __builtin_amdgcn_wmma_f32_16x16x4_f32(
          false, a, false, b, (short)0, acc[c], false, false);
    }
  }

  const int rbase = tile * 16 + 8 * hi;           // rows this lane-half owns
#pragma unroll
  for (int v = 0; v < 8; ++v) {
    int r = route[rbase + v];
    float val = acc[0][v];
    val = (r == 1) ? acc[1][v] : val;
    val = (r == 2) ? acc[2][v] : val;
    val = (r == 3) ? acc[3][v] : val;
    val = (r == 4) ? acc[4][v] : val;
    val = (r == 5) ? acc[5][v] : val;
    val = (r == 6) ? acc[6][v] : val;
    val = (r == 7) ? acc[7][v] : val;
    out[(size_t)(rbase + v) * N_CLASSES + cblock * 16 + lo] = val;
  }
}

// ---------------------------------------------------------------------------
extern "C" void kernel_launch(void* const* d_in, const int* in_sizes, int n_in,
                              void* d_out, int out_size, void* d_ws, size_t ws_size,
                              hipStream_t stream) {
  const float* h    = (const float*)d_in[0];     // [N, D]
  const int*   edge = (const int*)d_in[1];       // [2, E]
  const float* Ws   = (const float*)d_in[2];     // [C, D]
  const float* Wt   = (const float*)d_in[3];     // [C, O, D]
  float* out = (float*)d_out;                    // [N, O]

  char* ws = (char*)d_ws;
  float* sum   = (float*)ws;                                         // N*D f32 (becomes hm)
  float* cnt   = (float*)(ws + (size_t)N_NODES * DIM * sizeof(float));
  int*   route = (int*)  (ws + (size_t)N_NODES * DIM * sizeof(float)
                             + (size_t)N_NODES * sizeof(float));

  const int* esrc = edge;
  const int* edst = edge + N_EDGES;

  // 1) sum = h, cnt = 1 (self loops; also resets scratch each call)
  gppt_init<<<dim3((N_NODES * DIM / 4 + 255) / 256), dim3(256), 0, stream>>>(
      (const float4*)h, (float4*)sum, cnt);

  // 2) scatter-add edges (one wave32 per edge)
  gppt_scatter<<<dim3((N_EDGES * 32 + 255) / 256), dim3(256), 0, stream>>>(
      h, esrc, edst, sum, cnt);

  // 3) hm = sum/cnt, router argmax
  gppt_route<<<dim3(N_NODES / 8), dim3(256), 0, stream>>>(sum, cnt, Ws, route);

  // 4) expert einsum + hard routing, f32 WMMA
  gppt_expert<<<dim3(N_NODES / 16, 2), dim3(32, 4), 0, stream>>>(sum, Wt, route, out);
}